// OhemCrossEntropy2d_47485158425490
// MI455X (gfx1250) — compile-verified
//
#include <hip/hip_runtime.h>
#include <stdint.h>

// Problem geometry (fixed by the reference)
#define C_  19
#define H_  512
#define W_  1024
#define N_  8
#define HS_ 64
#define WS_ 128
#define HW_ ((size_t)H_ * (size_t)W_)
#define NPIX (N_ * H_ * W_)     // 4,194,304
#define NDS  (N_ * HS_ * WS_)   // 65,536
#define MIN_KEPT_DS 1562u       // 100000 / (8*8)
#define THRESH_ 0.7f
#define TILES_PER_BLOCK 4       // loss kernel: 4 x 256-pixel tiles per block

typedef __attribute__((ext_vector_type(2))) float    v2f;
typedef __attribute__((ext_vector_type(8))) float    v8f;
typedef __attribute__((ext_vector_type(4))) unsigned v4u;
typedef __attribute__((ext_vector_type(8))) int      v8i;
typedef __attribute__((ext_vector_type(4))) int      v4i;

// ---------------------------------------------------------------------------
// Wave32 sum-reduction through the matrix pipe (V_WMMA_F32_16X16X4_F32):
// A = lane values as 16x4 f32 (col 0 lanes 0-15, col 2 lanes 16-31), B = ones
// => D[i][j] = v[i]+v[i+16]; summing the 8 C/D VGPRs + one shfl_xor(16) gives
// the exact fp32 wave total in every lane. EXEC all-ones (uniform flow).
// ---------------------------------------------------------------------------
__device__ __forceinline__ float wave_sum_wmma(float v) {
    v2f a; a.x = v;    a.y = 0.0f;
    v2f b; b.x = 1.0f; b.y = 1.0f;
    v8f c = {};
    c = __builtin_amdgcn_wmma_f32_16x16x4_f32(
            /*neg_a=*/false, a, /*neg_b=*/false, b,
            /*c_mod=*/(short)0, c, /*reuse_a=*/false, /*reuse_b=*/false);
    float s = c[0] + c[1] + c[2] + c[3] + c[4] + c[5] + c[6] + c[7];
    s += __shfl_xor(s, 16, 32);
    return s;
}

// ---------------------------------------------------------------------------
// Tensor Data Mover: 2-D tile load Global->LDS.
//   tile: dim0 = 256 contiguous f32 (one row chunk), dim1 = 19 channels,
//   row stride = H*W elements. Descriptor bitfields per CDNA5 ISA
//   §8.3/8.4 (group0: count=1, lds_addr, global_addr, type=2;
//   group1: data_size=4B, tensor dims (large, in-bounds), tile dims, stride).
// Tracked by TENSORcnt; in-order per wave.
// Toolchain here is the 6-arg builtin:
//   (uint32x4 g0, int32x8 g1, int32x4, int32x4, int32x8, i32 cpol)
// ---------------------------------------------------------------------------
__device__ __forceinline__ void tdm_load_tile(const float* gptr, unsigned lds_byte_off) {
    unsigned long long ga = (unsigned long long)(uintptr_t)gptr;
    v4u g0;
    g0.x = 1u;                                            // count=1 (valid user D#)
    g0.y = lds_byte_off;                                  // lds_addr (bytes)
    g0.z = (unsigned)(ga & 0xFFFFFFFFu);                  // global_addr[31:0]
    g0.w = (unsigned)((ga >> 32) & 0x01FFFFFFu)           // global_addr[56:32]
         | 0x80000000u;                                   // type=2 ("image")
    // tensor_dim0 = 0x40000000 (huge, no OOB), tensor_dim1 = 0x1000,
    // tile_dim0 = 256, tile_dim1 = 19, tile_dim2 = 0,
    // tensor_dim0_stride = H*W = 524288, tensor_dim1_stride = 0.
    v8i g1 = { 0x00020000,                                // wg_mask=0, data_size=2 (4B)
               0x00000000,                                // bar_addr=0, td0[15:0]=0
               0x10004000,                                // td0[31:16]=0x4000, td1[15:0]=0x1000
               0x01000000,                                // td1 hi=0, tile_dim0=256
               19,                                        // tile_dim1=19, tile_dim2=0
               (int)(H_ * W_),                            // tensor_dim0_stride lo
               0, 0 };                                    // stride0 hi / stride1
    v4i gz4 = { 0, 0, 0, 0 };                             // groups 2/3 unused (2-D)
    v8i gz8 = { 0, 0, 0, 0, 0, 0, 0, 0 };                 // trailing group unused
    __builtin_amdgcn_tensor_load_to_lds(g0, g1, gz4, gz4, gz8, /*cpol=*/0);
}

// Softmax prob of channel `lbl` at one full-res pixel (x0 = &predict[n][0][h][w]).
__device__ __forceinline__ float softmax_prob_at(const float* __restrict__ x0, int lbl) {
    float x[C_];
    float m = -3.4e38f;
#pragma unroll
    for (int c = 0; c < C_; ++c) {
        x[c] = x0[(size_t)c * HW_];
        m = fmaxf(m, x[c]);
    }
    float s = 0.0f, xl = x[0];
#pragma unroll
    for (int c = 0; c < C_; ++c) {
        s += __expf(x[c] - m);
        if (c == lbl) xl = x[c];   // static c -> v_cndmask chain, no scratch
    }
    return __expf(xl - m) / s;
}

// ---------------------------------------------------------------------------
// Kernel 0: zero the accumulators (d_ws is poisoned, not re-zeroed by harness)
// ---------------------------------------------------------------------------
__global__ void init_kernel(float* sum_out, unsigned* nvalid) {
    *sum_out = 0.0f;
    *nvalid  = 0u;
}

// ---------------------------------------------------------------------------
// Kernel 1: downsampled pred keys + valid count.
// nearest label at (8o+4, 8o+4); 4-tap bilinear (weights 0.25) of softmax prob
// at rows {8o+3,8o+4} x cols {8o+3,8o+4} (sample center 8o+3.5).
// Key = float bits (non-negative floats sort as unsigned); invalid -> +inf.
// ---------------------------------------------------------------------------
__global__ void __launch_bounds__(256)
ds_kernel(const float* __restrict__ predict, const int* __restrict__ target,
          unsigned* __restrict__ keys, unsigned* __restrict__ nvalid) {
    int o  = blockIdx.x * 256 + threadIdx.x;       // exact grid: o < NDS
    int n  = o / (HS_ * WS_);
    int r  = o % (HS_ * WS_);
    int oh = r / WS_, ow = r % WS_;

    int lbl = target[((size_t)n * H_ + (oh * 8 + 4)) * W_ + (ow * 8 + 4)];
    bool valid = (lbl != 0);

    unsigned key = 0x7F800000u;                    // +inf: sorts last
    if (valid) {
        float acc = 0.0f;
#pragma unroll
        for (int t = 0; t < 4; ++t) {
            int th = oh * 8 + 3 + (t >> 1);
            int tw = ow * 8 + 3 + (t & 1);
            const float* x0 = predict + (size_t)n * C_ * HW_ + (size_t)th * W_ + tw;
            __builtin_prefetch(x0, 0, 0);          // global_prefetch_b8
            acc += softmax_prob_at(x0, lbl);
        }
        key = __float_as_uint(0.25f * acc);
    }
    keys[o] = key;

    unsigned long long ball = __ballot(valid);     // wave32: low 32 bits
    if ((threadIdx.x & 31) == 0)
        atomicAdd(nvalid, (unsigned)__popcll(ball));
}

// ---------------------------------------------------------------------------
// Kernel 2: k-th smallest via 4-pass (8 bits/pass) radix select over 65,536
// keys, single workgroup, LDS histogram. Then OHEM threshold rule.
// ---------------------------------------------------------------------------
__global__ void __launch_bounds__(1024)
radix_select_kernel(const unsigned* __restrict__ keys,
                    const unsigned* __restrict__ nvalid_p,
                    float* __restrict__ thr_out) {
    __shared__ unsigned hist[256];
    __shared__ unsigned s_prefix;
    __shared__ unsigned s_want;

    unsigned tid = threadIdx.x;
    if (tid == 0) {
        unsigned nv = *nvalid_p;
        unsigned kt = nv < MIN_KEPT_DS ? nv : MIN_KEPT_DS;   // min(num_valid, min_kept)
        int k0 = (int)kt - 1;
        if (k0 < 0) k0 = 0;
        if (k0 > NDS - 1) k0 = NDS - 1;
        s_want   = (unsigned)k0 + 1;                         // 1-based rank
        s_prefix = 0u;
    }
    __syncthreads();

    for (int p = 0; p < 4; ++p) {
        if (tid < 256) hist[tid] = 0u;
        __syncthreads();

        int shift = 24 - 8 * p;
        unsigned pref = s_prefix;
        for (int i = tid; i < NDS; i += 1024) {
            unsigned key = keys[i];
            bool match = (p == 0) || ((key >> (32 - 8 * p)) == pref);
            if (match) atomicAdd(&hist[(key >> shift) & 0xFFu], 1u);
        }
        __syncthreads();

        if (tid == 0) {
            unsigned want = s_want, cum = 0u, b = 0u;
            for (; b < 256u; ++b) {
                unsigned h = hist[b];
                if (cum + h >= want) break;
                cum += h;
            }
            if (b > 255u) b = 255u;
            s_want   = want - cum;
            s_prefix = (pref << 8) | b;
        }
        __syncthreads();
    }

    if (tid == 0) {
        unsigned nv = *nvalid_p;
        float kth = __uint_as_float(s_prefix);
        float thr = kth > THRESH_ ? kth : THRESH_;
        if (MIN_KEPT_DS >= nv) thr = 1.0f;
        *thr_out = thr;
    }
}

// ---------------------------------------------------------------------------
// Kernel 3: fused softmax / OHEM-select / NLL, TDM double-buffered.
// Each block handles 1024 contiguous pixels (one full W row; blocks never
// straddle images since (H*W) % 1024 == 0). Wave 0 drives the Tensor Data
// Mover: tile t+1 streams into LDS while all 8 waves compute tile t.
// ---------------------------------------------------------------------------
__global__ void __launch_bounds__(256)
loss_kernel(const float* __restrict__ predict, const int* __restrict__ target,
            const float* __restrict__ thr_p, float* __restrict__ sum_out) {
    __shared__ float tile[2][C_ * 256];    // 2 x 19 KB double buffer
    __shared__ float lsum[8];

    const int  tid       = threadIdx.x;
    const int  blockBase = blockIdx.x * (256 * TILES_PER_BLOCK);
    const int  n         = blockBase / (int)(H_ * W_);
    const int  hwBase    = blockBase % (int)(H_ * W_);
    const float* imgBase = predict + (size_t)n * C_ * HW_;
    const bool issuer    = (tid < 32);     // wave 0 drives the TDM

    if (issuer)
        tdm_load_tile(imgBase + hwBase, (unsigned)(uintptr_t)&tile[0][0]);

    float thr = *thr_p;
    float acc = 0.0f;

#pragma unroll
    for (int t = 0; t < TILES_PER_BLOCK; ++t) {
        if (issuer) {
            if (t + 1 < TILES_PER_BLOCK) {
                tdm_load_tile(imgBase + hwBase + (t + 1) * 256,
                              (unsigned)(uintptr_t)&tile[(t + 1) & 1][0]);
                __builtin_amdgcn_s_wait_tensorcnt((short)1);  // tile t done (in-order)
            } else {
                __builtin_amdgcn_s_wait_tensorcnt((short)0);
            }
        }
        __syncthreads();                               // tile t visible to all waves

        const float* xt  = &tile[t & 1][0];
        const int    pix = blockBase + t * 256 + tid;
        const int    lbl = target[pix];

        float x[C_];
        float m = -3.4e38f;
#pragma unroll
        for (int c = 0; c < C_; ++c) {
            x[c] = xt[c * 256 + tid];                  // conflict-free DS reads
            m = fmaxf(m, x[c]);
        }
        float s = 0.0f, xl = x[0];
#pragma unroll
        for (int c = 0; c < C_; ++c) {
            s += __expf(x[c] - m);
            if (c == lbl) xl = x[c];
        }
        float lse  = m + __logf(s);
        float pl   = __expf(xl - m) / s;
        bool  kept = (lbl != 0) && (pl <= thr);
        acc += kept ? (lse - xl) : (lse - x[0]);       // nll of new_target

        __syncthreads();                               // buffer free before overwrite
    }

    // wave32 reduction through the matrix pipe (uniform flow, EXEC all-ones)
    float wsum = wave_sum_wmma(acc);

    if ((tid & 31) == 0) lsum[tid >> 5] = wsum;
    __syncthreads();
    if (tid == 0) {
        float tsum = 0.0f;
#pragma unroll
        for (int i = 0; i < 8; ++i) tsum += lsum[i];
        atomicAdd(sum_out, tsum);
    }
}

// ---------------------------------------------------------------------------
// Kernel 4: mean
// ---------------------------------------------------------------------------
__global__ void finalize_kernel(const float* __restrict__ sum_in, float* __restrict__ out) {
    *out = (*sum_in) * (1.0f / (float)NPIX);
}

// ---------------------------------------------------------------------------
extern "C" void kernel_launch(void* const* d_in, const int* in_sizes, int n_in,
                              void* d_out, int out_size, void* d_ws, size_t ws_size,
                              hipStream_t stream) {
    const float* predict = (const float*)d_in[0];   // (8,19,512,1024) fp32
    const int*   target  = (const int*)d_in[1];     // (8,512,1024) int32
    float*       out     = (float*)d_out;           // scalar fp32

    // Workspace layout: [0]=sum f32, [1]=nvalid u32, [2]=thr f32, [3]=pad,
    //                   [4..] = 65,536 u32 keys (256 KB)
    float*    ws_sum    = (float*)d_ws;
    unsigned* ws_nvalid = (unsigned*)d_ws + 1;
    float*    ws_thr    = (float*)d_ws + 2;
    unsigned* ws_keys   = (unsigned*)d_ws + 4;

    init_kernel<<<1, 1, 0, stream>>>(ws_sum, ws_nvalid);
    ds_kernel<<<NDS / 256, 256, 0, stream>>>(predict, target, ws_keys, ws_nvalid);
    radix_select_kernel<<<1, 1024, 0, stream>>>(ws_keys, ws_nvalid, ws_thr);
    loss_kernel<<<NPIX / (256 * TILES_PER_BLOCK), 256, 0, stream>>>(predict, target, ws_thr, ws_sum);
    finalize_kernel<<<1, 1, 0, stream>>>(ws_sum, out);
}